// Resize_35613868819061
// MI455X (gfx1250) — compile-verified
//
#include <hip/hip_runtime.h>
#include <hip/hip_bf16.h>
#include <cstdint>

// 2x trilinear upsample [2,96,96,96,2] f32 -> [2,192,192,192,2] f32.
// Bandwidth-bound (~130 MB total traffic => ~5.6us floor @ 23.3 TB/s).
// Data path: CDNA5 Tensor Data Mover stages a 9x9x9-voxel (float2) input tile
// (incl. +1 halo) into LDS with one tensor_load_to_lds (wave-0 issues,
// s_wait_tensorcnt, workgroup barrier publishes); compute is trivial
// averaging; output is streamed with non-temporal 128-bit stores since it is
// write-once (87% of the traffic) and must not thrash the near caches.

typedef __attribute__((ext_vector_type(4))) unsigned int u32x4;
typedef __attribute__((ext_vector_type(8))) unsigned int u32x8;
typedef __attribute__((ext_vector_type(4))) float        f32x4;

__device__ __forceinline__ int imin(int a, int b) { return a < b ? a : b; }
__device__ __forceinline__ float2 favg(float2 a, float2 b) {
    return make_float2(0.5f * (a.x + b.x), 0.5f * (a.y + b.y));
}

namespace {
constexpr int IN_D  = 96;
constexpr int OUT_D = 192;
constexpr int TI    = 8;   // input tile edge
constexpr int THL   = 9;   // input tile edge + halo
constexpr int TO    = 16;  // output tile edge
}

__global__ __launch_bounds__(256) void resize_zoom2_kernel(
    const float* __restrict__ vol, float* __restrict__ out) {
    __shared__ float2 tile[THL * THL * THL];  // 729 voxels (2ch) = 5832 B

    const int tx  = blockIdx.x;        // 0..11
    const int ty  = blockIdx.y;        // 0..11
    const int tzb = blockIdx.z;        // 0..23
    const int b   = tzb / 12;
    const int tz  = tzb % 12;

    // ---------------- TDM descriptor (wave-uniform -> SGPRs) ----------------
    // Tensor viewed as 3D float32 tensor [z=96][y=96][x*c=192], element = 4B.
    // Tile = [9 z-planes][9 y-rows][18 floats]; OOB reads zero-fill (never read).
    const uint64_t gbase =
        (uint64_t)(uintptr_t)vol +
        (uint64_t)b * ((uint64_t)IN_D * IN_D * IN_D * 2u * 4u) +
        (uint64_t)(((tz * TI) * IN_D + (ty * TI)) * IN_D + (tx * TI)) * 2u * 4u;
    const uint32_t lds_base = (uint32_t)(uintptr_t)(&tile[0]);

    u32x4 g0;
    g0[0] = 1u;                                    // count=1, user descriptor
    g0[1] = lds_base;                              // lds_addr (bytes)
    g0[2] = (uint32_t)gbase;                       // global_addr[31:0]
    g0[3] = (uint32_t)(gbase >> 32) | (2u << 30);  // global_addr[56:32] | type=2

    constexpr uint32_t S0 = 2u * IN_D;               // dim0 stride = 192 floats
    constexpr uint32_t S1 = 2u * IN_D * IN_D;        // dim1 stride = 18432 floats
    u32x8 g1;
    g1[0] = (2u << 16);                              // data_size=2 (4B); no mask/pad/iter
    g1[1] = ((uint32_t)(2 * IN_D)) << 16;            // tensor_dim0 = 192 (bits 63:48)
    g1[2] = ((uint32_t)IN_D) << 16;                  // tensor_dim1 = 96  (bits 95:80)
    g1[3] = ((uint32_t)(2 * THL)) << 16;             // tile_dim0 = 18    (bits 127:112)
    g1[4] = (uint32_t)THL | ((uint32_t)THL << 16);   // tile_dim1 = 9, tile_dim2 = 9
    g1[5] = S0;                                      // tensor_dim0_stride[31:0]
    g1[6] = (S1 & 0xFFFFu) << 16;                    // dim0_stride[47:32]=0 | dim1_stride[15:0]
    g1[7] = S1 >> 16;                                // tensor_dim1_stride[47:16]

    u32x4 g2; g2[0] = (uint32_t)IN_D; g2[1] = 0u; g2[2] = 0u; g2[3] = 0u;  // tensor_dim2=96
    u32x4 g3; g3[0] = 0u; g3[1] = 0u; g3[2] = 0u; g3[3] = 0u;

    if ((threadIdx.x >> 5) == 0) {  // wave 0 only: TDM ignores EXEC, so gate by wave
        asm volatile("tensor_load_to_lds %0, %1, %2, %3"
                     :: "s"(g0), "s"(g1), "s"(g2), "s"(g3)
                     : "memory");
        __builtin_amdgcn_s_wait_tensorcnt(0);
    }
    __syncthreads();

    // Clamp local halo index so boundary tiles reproduce the reference's clip
    // semantics (out[191] = in[95]) and never touch zero-filled OOB LDS slots.
    const int lmx = imin(TI, (IN_D - 1) - tx * TI);
    const int lmy = imin(TI, (IN_D - 1) - ty * TI);
    const int lmz = imin(TI, (IN_D - 1) - tz * TI);

    float* outp = out + (size_t)b * ((size_t)OUT_D * OUT_D * OUT_D * 2u);

    for (int m = threadIdx.x; m < TI * TI * TI; m += 256) {
        const int iz = m >> 6, iy = (m >> 3) & 7, ix = m & 7;
        const int x1 = imin(ix + 1, lmx);
        const int y1 = imin(iy + 1, lmy);
        const int z1 = imin(iz + 1, lmz);

        const float2 c000 = tile[(iz * THL + iy) * THL + ix];
        const float2 c001 = tile[(iz * THL + iy) * THL + x1];
        const float2 c010 = tile[(iz * THL + y1) * THL + ix];
        const float2 c011 = tile[(iz * THL + y1) * THL + x1];
        const float2 c100 = tile[(z1 * THL + iy) * THL + ix];
        const float2 c101 = tile[(z1 * THL + iy) * THL + x1];
        const float2 c110 = tile[(z1 * THL + y1) * THL + ix];
        const float2 c111 = tile[(z1 * THL + y1) * THL + x1];

        // x-odd lerps
        const float2 e00 = favg(c000, c001);
        const float2 e01 = favg(c010, c011);
        const float2 e10 = favg(c100, c101);
        const float2 e11 = favg(c110, c111);

        // (z-parity, y-parity) rows, each holding (x-even, x-odd)
        const float2 r00_0 = c000,               r00_1 = e00;
        const float2 r01_0 = favg(c000, c010),   r01_1 = favg(e00, e01);
        const float2 r10_0 = favg(c000, c100),   r10_1 = favg(e00, e10);
        const float2 r11_0 = favg(r01_0, favg(c100, c110));
        const float2 r11_1 = favg(r01_1, favg(e10, e11));

        const int ozg = tz * TO + 2 * iz;
        const int oyg = ty * TO + 2 * iy;
        const int oxg = tx * TO + 2 * ix;
        const size_t rowbase = (((size_t)ozg * OUT_D + oyg) * OUT_D + oxg) * 2u;
        const size_t dy = (size_t)OUT_D * 2u;
        const size_t dz = (size_t)OUT_D * OUT_D * 2u;

        f32x4 q;
        q.x = r00_0.x; q.y = r00_0.y; q.z = r00_1.x; q.w = r00_1.y;
        __builtin_nontemporal_store(q, reinterpret_cast<f32x4*>(outp + rowbase));
        q.x = r01_0.x; q.y = r01_0.y; q.z = r01_1.x; q.w = r01_1.y;
        __builtin_nontemporal_store(q, reinterpret_cast<f32x4*>(outp + rowbase + dy));
        q.x = r10_0.x; q.y = r10_0.y; q.z = r10_1.x; q.w = r10_1.y;
        __builtin_nontemporal_store(q, reinterpret_cast<f32x4*>(outp + rowbase + dz));
        q.x = r11_0.x; q.y = r11_0.y; q.z = r11_1.x; q.w = r11_1.y;
        __builtin_nontemporal_store(q, reinterpret_cast<f32x4*>(outp + rowbase + dz + dy));
    }
}

extern "C" void kernel_launch(void* const* d_in, const int* in_sizes, int n_in,
                              void* d_out, int out_size, void* d_ws, size_t ws_size,
                              hipStream_t stream) {
    (void)in_sizes; (void)n_in; (void)out_size; (void)d_ws; (void)ws_size;
    const float* vol = (const float*)d_in[0];
    float* out = (float*)d_out;
    dim3 grid(12, 12, 24);   // 12^3 tiles of 16^3 outputs, x batch=2 folded into z
    resize_zoom2_kernel<<<grid, dim3(256, 1, 1), 0, stream>>>(vol, out);
}